// MeanAggregator_72980084293969
// MI455X (gfx1250) — compile-verified
//
#include <hip/hip_runtime.h>
#include <stdint.h>

#define AS1 __attribute__((address_space(1)))
#define AS3 __attribute__((address_space(3)))

typedef __attribute__((ext_vector_type(2))) float v2f;
typedef __attribute__((ext_vector_type(8))) float v8f;

// ---------------- Pass 1: zero degree scratch ----------------
__global__ __launch_bounds__(256) void k_zero_deg(unsigned* __restrict__ deg, int n) {
    int i = blockIdx.x * 256 + threadIdx.x;
    if (i < n) deg[i] = 0u;
}

// ---------------- Pass 2: degree count over dst ----------------
__global__ __launch_bounds__(256) void k_count_deg(const int* __restrict__ edge_dst,
                                                   unsigned* __restrict__ deg, int e_cnt) {
    int e = blockIdx.x * 256 + threadIdx.x;
    if (e < e_cnt) atomicAdd(&deg[edge_dst[e]], 1u);
}

// ---------------- Pass 3: out = diag(1/(1+deg)) * F  via WMMA f32 16x16x4 ----
// Also writes inv[i] = rsqrt(1 + deg[i]) for the scatter pass.
// One wave handles 16 nodes (two 16x16 feature tiles, since D=32).
// A (16x4 f32) layout: lane l holds row M=l&15; VGPR0 -> K = 2*(l>>4), VGPR1 -> K+1.
// B (4x16 f32) layout: lane l holds col N=l&15; VGPR0 -> K = 2*(l>>4), VGPR1 -> K+1.
// C/D (16x16 f32):     lane l holds col N=l&15; VGPR v -> M = v + 8*(l>>4).
__global__ __launch_bounds__(256) void k_init_wmma(const float* __restrict__ feat,
                                                   const unsigned* __restrict__ deg,
                                                   float* __restrict__ inv,
                                                   float* __restrict__ out, int n) {
    const unsigned un     = (unsigned)n;
    const unsigned wave   = (blockIdx.x * 256u + threadIdx.x) >> 5;
    const unsigned nwaves = (un + 15u) / 16u;
    if (wave >= nwaves) return;                       // wave-uniform: EXEC stays full for WMMA
    const unsigned i0   = wave * 16u;
    const unsigned lane = threadIdx.x & 31u;
    const unsigned M    = lane & 15u;
    const unsigned hi   = lane >> 4;                  // lane half selects K pair / M+8
    const unsigned node = i0 + M;
    const unsigned nodeC = node < un ? node : un - 1u;
    float dp1 = 1.0f + (float)deg[nodeC];
    float w2  = __builtin_amdgcn_rcpf(dp1);           // self-loop weight 1/(1+deg)
    if (!hi && node < un) inv[node] = rsqrtf(dp1);
    const unsigned k0   = hi * 2u;
    const bool     full = (i0 + 16u <= un);           // wave-uniform
    for (unsigned c0 = 0; c0 < 32; c0 += 16) {
        v8f acc = {};
#pragma unroll
        for (unsigned g = 0; g < 4; ++g) {
            unsigned r0 = i0 + 4u * g + k0;           // feature rows this lane supplies to B
            unsigned rA = r0      < un ? r0      : un - 1u;   // clamp: branch-free loads
            unsigned rB = r0 + 1u < un ? r0 + 1u : un - 1u;
            v2f a, b;
            a.x = (M == 4u * g + k0)      ? w2 : 0.0f;
            a.y = (M == 4u * g + k0 + 1u) ? w2 : 0.0f;
            float bx = feat[(size_t)rA * 32 + c0 + M];
            float by = feat[(size_t)rB * 32 + c0 + M];
            b.x = (r0      < un) ? bx : 0.0f;         // v_cndmask, no exec juggling
            b.y = (r0 + 1u < un) ? by : 0.0f;
            acc = __builtin_amdgcn_wmma_f32_16x16x4_f32(false, a, false, b,
                                                        (short)0, acc, false, false);
        }
        if (full) {                                   // hot path: straight-line stores
#pragma unroll
            for (unsigned v = 0; v < 8; ++v)
                out[(size_t)(i0 + v + 8u * hi) * 32 + c0 + M] = acc[v];
        } else {                                      // rare tail wave only
#pragma unroll
            for (unsigned v = 0; v < 8; ++v) {
                unsigned row = i0 + v + 8u * hi;
                if (row < un) out[(size_t)row * 32 + c0 + M] = acc[v];
            }
        }
    }
}

// ---------------- Pass 4: edge scatter with async-LDS index staging ----------
// 256 threads stage 256 edges' src/dst into LDS asynchronously, precompute the
// per-edge weight w = inv[src]*inv[dst] once per edge, then each thread handles
// 8 edge-quarters (float4 gather + 4 f32 atomic adds).
#define ASYNC_CAP (__has_builtin(__builtin_amdgcn_global_load_async_to_lds_b32))

__global__ __launch_bounds__(256) void k_scatter(const float4* __restrict__ feat4,
                                                 const int* __restrict__ edge_src,
                                                 const int* __restrict__ edge_dst,
                                                 const float* __restrict__ inv,
                                                 float* __restrict__ out, int e_cnt) {
    __shared__ int   s_src[256];
    __shared__ int   s_dst[256];
    __shared__ float s_w[256];
    const unsigned t    = threadIdx.x;
    const unsigned base = blockIdx.x * 256u;
    unsigned g_idx = base + t;
    if (g_idx >= (unsigned)e_cnt) g_idx = (unsigned)e_cnt - 1u;  // clamp: keep addresses valid
#if ASYNC_CAP
    {
        // builtin signature is typed: (as1 int*, as3 int*, imm offset, imm cpol)
        AS1 int* gsrc = (AS1 int*)(uintptr_t)(edge_src + g_idx);
        AS1 int* gdst = (AS1 int*)(uintptr_t)(edge_dst + g_idx);
        AS3 int* lsrc = (AS3 int*)(unsigned)(uintptr_t)&s_src[t];
        AS3 int* ldst = (AS3 int*)(unsigned)(uintptr_t)&s_dst[t];
        __builtin_amdgcn_global_load_async_to_lds_b32(gsrc, lsrc, 0, 0);
        __builtin_amdgcn_global_load_async_to_lds_b32(gdst, ldst, 0, 0);
    }
  #if __has_builtin(__builtin_amdgcn_s_wait_asynccnt)
    __builtin_amdgcn_s_wait_asynccnt(0);
  #else
    asm volatile("s_wait_asynccnt 0" ::: "memory");
  #endif
#else
  #warning "gfx1250 async-to-LDS builtin unavailable: falling back to direct LDS stores"
    s_src[t] = edge_src[g_idx];
    s_dst[t] = edge_dst[g_idx];
#endif
    __syncthreads();
    // per-edge weight once (instead of 8 redundant inv[] gathers per edge)
    s_w[t] = inv[s_src[t]] * inv[s_dst[t]];
    __syncthreads();

    const unsigned q  = t & 7u;       // which float4 quarter of the 32-float row
    const unsigned el = t >> 3;       // edge slot 0..31 within sub-chunk
#pragma unroll 2
    for (unsigned sub = 0; sub < 8; ++sub) {
        unsigned e = el + 32u * sub;
        if (base + e >= (unsigned)e_cnt) continue;
        int   s = s_src[e];
        int   d = s_dst[e];
        float w = s_w[e];
        if (sub + 1u < 8u && base + e + 32u < (unsigned)e_cnt) {
            int sn = s_src[e + 32u];                       // prefetch next gather row
            __builtin_prefetch(&feat4[(size_t)sn * 8 + q], 0, 0);
        }
        float4 f = feat4[(size_t)s * 8 + q];               // GLOBAL_LOAD_B128 gather
        float* o = out + (size_t)d * 32 + (size_t)q * 4;
        atomicAdd(o + 0, w * f.x);
        atomicAdd(o + 1, w * f.y);
        atomicAdd(o + 2, w * f.z);
        atomicAdd(o + 3, w * f.w);
    }
}

// ---------------- Launcher ----------------
extern "C" void kernel_launch(void* const* d_in, const int* in_sizes, int n_in,
                              void* d_out, int out_size, void* d_ws, size_t ws_size,
                              hipStream_t stream) {
    const float* feat = (const float*)d_in[0];
    const int*   esrc = (const int*)d_in[1];
    const int*   edst = (const int*)d_in[2];
    float*       out  = (float*)d_out;

    const int n = in_sizes[0] / 32;   // N_NODES
    const int e = in_sizes[1];        // N_EDGES

    unsigned* deg = (unsigned*)d_ws;
    float*    inv = (float*)((char*)d_ws + (((size_t)n * 4 + 255) & ~(size_t)255));

    k_zero_deg <<<(n + 255) / 256, 256, 0, stream>>>(deg, n);
    k_count_deg<<<(e + 255) / 256, 256, 0, stream>>>(edst, deg, e);

    const int nwaves = (n + 15) / 16;                  // one wave per 16 nodes
    k_init_wmma<<<(nwaves * 32 + 255) / 256, 256, 0, stream>>>(feat, deg, inv, out, n);

    k_scatter  <<<(e + 255) / 256, 256, 0, stream>>>((const float4*)feat, esrc, edst, inv, out, e);
}